// TacrolimusPK_5961414607271
// MI455X (gfx1250) — compile-verified
//
#include <hip/hip_runtime.h>

// Memory-bound PK ODE RHS for MI455X (gfx1250, wave32).
// Data path: per block, one wave issues 11 Tensor-Data-Mover descriptors that
// DMA 4KB contiguous rows of each input stream into LDS (TENSORcnt-tracked),
// then all 8 waves compute from LDS and stream results out with NT b128
// stores (protects L2 residency of the 176MB input set across graph replays).

typedef float        v4f  __attribute__((ext_vector_type(4)));
typedef unsigned int u32x4 __attribute__((ext_vector_type(4)));
typedef int          i32x4 __attribute__((ext_vector_type(4)));
typedef int          i32x8 __attribute__((ext_vector_type(8)));

#define CHUNK 1024   // floats per array per block (4KB per TDM descriptor)
#define BLOCK 256    // 8 wave32 waves; 256 lanes * 4 floats = CHUNK

#if __has_builtin(__builtin_amdgcn_tensor_load_to_lds) && \
    __has_builtin(__builtin_amdgcn_s_wait_tensorcnt)
#define USE_TDM 1
#else
#define USE_TDM 0
#endif

__global__ __launch_bounds__(BLOCK, 1) void pk_rhs_tdm_kernel(
    const float* __restrict__ a_depot,
    const float* __restrict__ a_gut1,
    const float* __restrict__ a_gut2,
    const float* __restrict__ a_gut3,
    const float* __restrict__ a_central,
    const float* __restrict__ a_periph,
    const float* __restrict__ p_ktr,
    const float* __restrict__ p_cl,
    const float* __restrict__ p_q,
    const float* __restrict__ p_vc,
    const float* __restrict__ p_vp,
    float* __restrict__ out,
    long long n)
{
    __shared__ __align__(16) float smem[11][CHUNK];   // 44KB of the 320KB/WGP pool

    const long long blockStart = (long long)blockIdx.x * CHUNK;
    const int tid = threadIdx.x;

    float* __restrict__ o0 = out;
    float* __restrict__ o1 = out + n;
    float* __restrict__ o2 = out + 2 * n;
    float* __restrict__ o3 = out + 3 * n;
    float* __restrict__ o4 = out + 4 * n;
    float* __restrict__ o5 = out + 5 * n;

    const float* __restrict__ src[11] = {
        a_depot, a_gut1, a_gut2, a_gut3, a_central, a_periph,
        p_ktr, p_cl, p_q, p_vc, p_vp };

    if (blockStart + CHUNK <= n) {
        // ---------------- full chunk: TDM global->LDS staging ----------------
#if USE_TDM
        if (tid < 32) {   // wave 0 issues all input DMA for the block
#pragma unroll
            for (int a = 0; a < 11; ++a) {
                const unsigned lds_addr =
                    (unsigned)(unsigned long long)(const void*)&smem[a][0];
                const unsigned long long ga =
                    (unsigned long long)(const void*)(src[a] + blockStart);

                // D# group 0: count=1 | lds_addr | global_addr[56:0] | type=2
                u32x4 g0;
                g0[0] = 1u;
                g0[1] = lds_addr;
                g0[2] = (unsigned)ga;
                g0[3] = ((unsigned)((ga >> 32) & 0x01FFFFFFull)) | 0x80000000u;

                // D# group 1: data_size=4B; tensor_dim0=tile_dim0=CHUNK (1 row)
                i32x8 g1;
                g1[0] = 0x00020000;           // workgroup_mask=0, data_size=2 (4B)
                g1[1] = (int)(CHUNK << 16);   // tensor_dim0[15:0] at bits 63:48
                g1[2] = (int)(1u << 16);      // tensor_dim1 = 1
                g1[3] = (int)(CHUNK << 16);   // tile_dim0 = CHUNK
                g1[4] = 1;                    // tile_dim1 = 1, tile_dim2 = 0
                g1[5] = CHUNK;                // tensor_dim0_stride[31:0]
                g1[6] = 0;                    // stride hi / tensor_dim1_stride lo
                g1[7] = 0;

                // D# groups 2/3: dims 2..4 = 1 (index 0 in bounds), tiles unused
                i32x4 g2;
                g2[0] = 1;                    // tensor_dim2 = 1
                g2[1] = 1;                    // tensor_dim3 = 1
                g2[2] = CHUNK;                // tensor_dim2_stride[31:0]
                g2[3] = 0;                    // stride hi, tile_dim3 = 0
                i32x4 g3;
                g3[0] = CHUNK;                // tensor_dim3_stride[31:0]
                g3[1] = (int)(1u << 16);      // tensor_dim4 = 1
                g3[2] = 0;                    // tile_dim4 = 0
                g3[3] = 0;

#if defined(__clang_major__) && (__clang_major__ >= 23)
                i32x8 g4 = {0, 0, 0, 0, 0, 0, 0, 0};
                __builtin_amdgcn_tensor_load_to_lds(g0, g1, g2, g3, g4, 0);
#else
                __builtin_amdgcn_tensor_load_to_lds(g0, g1, g2, g3, 0);
#endif
            }
            __builtin_amdgcn_s_wait_tensorcnt(0);   // all 11 DMAs landed in LDS
        }
        __syncthreads();

        const int li = tid * 4;                      // this lane's floats in LDS
        v4f ad  = *(const v4f*)&smem[0][li];
        v4f g1v = *(const v4f*)&smem[1][li];
        v4f g2v = *(const v4f*)&smem[2][li];
        v4f g3v = *(const v4f*)&smem[3][li];
        v4f ac  = *(const v4f*)&smem[4][li];
        v4f ap  = *(const v4f*)&smem[5][li];
        v4f ktr = *(const v4f*)&smem[6][li];
        v4f cl  = *(const v4f*)&smem[7][li];
        v4f q   = *(const v4f*)&smem[8][li];
        v4f vc  = *(const v4f*)&smem[9][li];
        v4f vp  = *(const v4f*)&smem[10][li];
#else
        // fallback: direct b128 loads
        const long long gi0 = blockStart + tid * 4;
        v4f ad  = *(const v4f*)(a_depot   + gi0);
        v4f g1v = *(const v4f*)(a_gut1    + gi0);
        v4f g2v = *(const v4f*)(a_gut2    + gi0);
        v4f g3v = *(const v4f*)(a_gut3    + gi0);
        v4f ac  = *(const v4f*)(a_central + gi0);
        v4f ap  = *(const v4f*)(a_periph  + gi0);
        v4f ktr = *(const v4f*)(p_ktr + gi0);
        v4f cl  = *(const v4f*)(p_cl  + gi0);
        v4f q   = *(const v4f*)(p_q   + gi0);
        v4f vc  = *(const v4f*)(p_vc  + gi0);
        v4f vp  = *(const v4f*)(p_vp  + gi0);
        const int li = tid * 4;
#endif

        v4f kel = cl / vc;
        v4f k12 = q  / vc;
        v4f k21 = q  / vp;

        v4f d0 = -(ktr * ad);
        v4f d1 = ktr * ad  - ktr * g1v;
        v4f d2 = ktr * g1v - ktr * g2v;
        v4f d3 = ktr * g2v - ktr * g3v;
        v4f d4 = ktr * g3v - kel * ac - k12 * ac + k21 * ap;
        v4f d5 = k12 * ac  - k21 * ap;

        const long long gi = blockStart + li;
        __builtin_nontemporal_store(d0, (v4f*)(o0 + gi));
        __builtin_nontemporal_store(d1, (v4f*)(o1 + gi));
        __builtin_nontemporal_store(d2, (v4f*)(o2 + gi));
        __builtin_nontemporal_store(d3, (v4f*)(o3 + gi));
        __builtin_nontemporal_store(d4, (v4f*)(o4 + gi));
        __builtin_nontemporal_store(d5, (v4f*)(o5 + gi));
    } else {
        // ---------------- partial tail chunk: direct scalar path ----------------
        for (long long k = blockStart + tid; k < n; k += BLOCK) {
            float ad  = a_depot[k],   tg1 = a_gut1[k], tg2 = a_gut2[k], tg3 = a_gut3[k];
            float ac  = a_central[k], ap  = a_periph[k];
            float ktr = p_ktr[k], cl = p_cl[k], q = p_q[k], vc = p_vc[k], vp = p_vp[k];

            float kel = cl / vc;
            float k12 = q  / vc;
            float k21 = q  / vp;

            o0[k] = -(ktr * ad);
            o1[k] = ktr * ad  - ktr * tg1;
            o2[k] = ktr * tg1 - ktr * tg2;
            o3[k] = ktr * tg2 - ktr * tg3;
            o4[k] = ktr * tg3 - kel * ac - k12 * ac + k21 * ap;
            o5[k] = k12 * ac  - k21 * ap;
        }
    }
}

extern "C" void kernel_launch(void* const* d_in, const int* in_sizes, int n_in,
                              void* d_out, int out_size, void* d_ws, size_t ws_size,
                              hipStream_t stream) {
    // setup_inputs() order:
    // 0:t  1:A_depot 2:A_gut1 3:A_gut2 4:A_gut3 5:A_central 6:A_peripheral
    // 7:Ktr 8:CL 9:Q 10:Vc 11:Vp
    const float* a_depot   = (const float*)d_in[1];
    const float* a_gut1    = (const float*)d_in[2];
    const float* a_gut2    = (const float*)d_in[3];
    const float* a_gut3    = (const float*)d_in[4];
    const float* a_central = (const float*)d_in[5];
    const float* a_periph  = (const float*)d_in[6];
    const float* p_ktr     = (const float*)d_in[7];
    const float* p_cl      = (const float*)d_in[8];
    const float* p_q       = (const float*)d_in[9];
    const float* p_vc      = (const float*)d_in[10];
    const float* p_vp      = (const float*)d_in[11];

    const long long n = in_sizes[1];
    int grid = (int)((n + CHUNK - 1) / CHUNK);
    if (grid < 1) grid = 1;

    pk_rhs_tdm_kernel<<<grid, BLOCK, 0, stream>>>(
        a_depot, a_gut1, a_gut2, a_gut3, a_central, a_periph,
        p_ktr, p_cl, p_q, p_vc, p_vp,
        (float*)d_out, n);
}